// CensusLoss_71451075936664
// MI455X (gfx1250) — compile-verified
//
#include <hip/hip_runtime.h>

#define TILE  8
#define HALO  3
#define LDIM  (TILE + 2*HALO)        // 14
#define LVOL  (LDIM*LDIM*LDIM)       // 2744
#define DVOL  64
#define NVOX  (DVOL*DVOL*DVOL)       // 262144
#define NBLOCKS 1024                 // 8*8*8 tiles * N=2

typedef __attribute__((ext_vector_type(2))) float v2f;
typedef __attribute__((ext_vector_type(8))) float v8f;

// Stage 1: per-block masked partial sums of mean_c( d/(0.1+d) ).
__global__ __launch_bounds__(512) void census_partial(
    const float* __restrict__ img0, const float* __restrict__ img1,
    float* __restrict__ blockSums)
{
  __shared__ float2 tile[LVOL];      // .x = img0, .y = img1  (ds_load_b64 per tap)

  const int n  = blockIdx.z >> 3;
  const int bz = (blockIdx.z & 7) * TILE;
  const int by = blockIdx.y * TILE;
  const int bx = blockIdx.x * TILE;
  const int tid = (threadIdx.z * TILE + threadIdx.y) * TILE + threadIdx.x;

  const float* __restrict__ b0 = img0 + n * NVOX;
  const float* __restrict__ b1 = img1 + n * NVOX;

  if (tid == 0) {  // global_prefetch_b8 on the tile base lines
    __builtin_prefetch(b0 + (bz*DVOL + by)*DVOL + bx, 0, 1);
    __builtin_prefetch(b1 + (bz*DVOL + by)*DVOL + bx, 0, 1);
  }

  // Stage 14^3 halo'd tile of both images into LDS.
  // Interior: async global->LDS DMA path (ASYNCcnt). Border: zero-fill via DS.
  for (int i = tid; i < LVOL; i += 512) {
    int lz = i / (LDIM*LDIM);
    int r  = i - lz * (LDIM*LDIM);
    int ly = r / LDIM;
    int lx = r - ly * LDIM;
    int gz = bz - HALO + lz, gy = by - HALO + ly, gx = bx - HALO + lx;
    if ((unsigned)gz < DVOL && (unsigned)gy < DVOL && (unsigned)gx < DVOL) {
      // Low 32 bits of the flat (generic) address of an LDS location are the
      // LDS byte offset (aperture high half selects LDS).
      unsigned lo = (unsigned)(uintptr_t)(tile + i);
      unsigned go = (unsigned)(((gz*DVOL + gy)*DVOL + gx) * 4);
      asm volatile("global_load_async_to_lds_b32 %0, %1, %2"
                   :: "v"(lo),      "v"(go), "s"(b0) : "memory");
      asm volatile("global_load_async_to_lds_b32 %0, %1, %2"
                   :: "v"(lo + 4u), "v"(go), "s"(b1) : "memory");
    } else {
      tile[i] = make_float2(0.f, 0.f);
    }
  }
  asm volatile("s_wait_asynccnt 0" ::: "memory");  // drain async->LDS before barrier
  __syncthreads();

  const float2 c = tile[((threadIdx.z+HALO)*LDIM + threadIdx.y+HALO)*LDIM + threadIdx.x+HALO];

  float acc = 0.f;
  for (int dz = 0; dz < 7; ++dz) {
#pragma unroll
    for (int dy = 0; dy < 7; ++dy) {
      int rowbase = ((threadIdx.z + dz)*LDIM + (threadIdx.y + dy))*LDIM + threadIdx.x;
#pragma unroll
      for (int dx = 0; dx < 7; ++dx) {
        float2 nb = tile[rowbase + dx];
        float t0  = nb.x - c.x;
        float t1  = nb.y - c.y;
        float t0n = t0 * __builtin_amdgcn_rsqf(__builtin_fmaf(t0, t0, 0.81f));
        float t1n = t1 * __builtin_amdgcn_rsqf(__builtin_fmaf(t1, t1, 0.81f));
        float df  = t0n - t1n;
        float d   = df * df;
        acc = __builtin_fmaf(d, __builtin_amdgcn_rcpf(d + 0.1f), acc);
      }
    }
  }

  // valid_mask: zero on the 1-voxel volume border
  int gz = bz + threadIdx.z, gy = by + threadIdx.y, gx = bx + threadIdx.x;
  bool interior = (gz >= 1) & (gz <= DVOL-2) & (gy >= 1) & (gy <= DVOL-2)
                & (gx >= 1) & (gx <= DVOL-2);
  float partial = interior ? acc * (1.0f / 343.0f) : 0.f;

  // Deterministic block tree-reduce (reuse tile LDS).
  __syncthreads();
  float* red = reinterpret_cast<float*>(tile);
  red[tid] = partial;
  __syncthreads();
  for (int s = 256; s > 0; s >>= 1) {
    if (tid < s) red[tid] += red[tid + s];
    __syncthreads();
  }
  if (tid == 0) {
    int blin = (blockIdx.z * gridDim.y + blockIdx.y) * gridDim.x + blockIdx.x;
    blockSums[blin] = red[0];
  }
}

// Stage 2: one wave reduces the 1024 block sums; final 32->1 via f32 WMMA.
// A(16x4): lane m holds A[m][0]=p_m (lanes 0-15), lane 16+m holds A[m][2]=p_{16+m};
// B = ones(4x16)  =>  D[m][n] = p_m + p_{m+16} for every n.
__global__ __launch_bounds__(32) void census_finalize(
    const float* __restrict__ blockSums, float* __restrict__ out)
{
  int lane = threadIdx.x;
  float s = 0.f;
  for (int i = lane; i < NBLOCKS; i += 32) s += blockSums[i];

  v2f a; a[0] = s;   a[1] = 0.f;
  v2f b; b[0] = 1.f; b[1] = 1.f;
  v8f cacc = {};
  cacc = __builtin_amdgcn_wmma_f32_16x16x4_f32(
      /*neg_a=*/false, a, /*neg_b=*/false, b,
      /*c_mod=*/(short)0, cacc, /*reuse_a=*/false, /*reuse_b=*/false);

  // lane0 holds D[0..7][0] in cacc[0..7]; lane16 holds D[8..15][0].
  float rs = cacc[0]+cacc[1]+cacc[2]+cacc[3]+cacc[4]+cacc[5]+cacc[6]+cacc[7];
  float other = __shfl(rs, 16, 32);
  if (lane == 0) out[0] = (rs + other) * (1.0f / 524288.0f);  // / (N*D*H*W)
}

extern "C" void kernel_launch(void* const* d_in, const int* in_sizes, int n_in,
                              void* d_out, int out_size, void* d_ws, size_t ws_size,
                              hipStream_t stream) {
  const float* img0 = (const float*)d_in[0];
  const float* img1 = (const float*)d_in[1];
  float* out   = (float*)d_out;
  float* bsums = (float*)d_ws;                 // 1024 floats, fully rewritten each call

  dim3 grid(8, 8, 16);                         // 8x8x8 tiles, z also encodes N=2
  dim3 block(TILE, TILE, TILE);                // 512 threads = 16 waves
  census_partial <<<grid, block, 0, stream>>>(img0, img1, bsums);
  census_finalize<<<1, 32, 0, stream>>>(bsums, out);
}